// MeanAggregator_26268019983003
// MI455X (gfx1250) — compile-verified
//
#include <hip/hip_runtime.h>

typedef __attribute__((ext_vector_type(2))) float v2f;
typedef __attribute__((ext_vector_type(8))) float v8f;

#define DCOLS 64

// One wave32 per segment. segment_ids are sorted, so each segment is a
// contiguous run [lo, hi). Column sums are computed on the matrix pipe:
//   A = ones(16x4)  =>  D[m,n] = sum_k B[k,n]   (V_WMMA_F32_16X16X4_F32)
// Each WMMA sums 4 rows x 16 columns; 4 tiles cover D=64 columns.
__global__ __launch_bounds__(32) void seg_mean_wmma_kernel(
    const float* __restrict__ values,
    const long long* __restrict__ seg_ids,
    float* __restrict__ out,
    int E, int N)
{
    const int seg = blockIdx.x;
    if (seg >= N) return;

    // --- uniform binary searches (whole wave identical -> scalarized) ---
    const long long t0 = (long long)seg;
    const long long t1 = t0 + 1;
    int lo, hi;
    {
        int l = 0, r = E;
        while (l < r) { int m = (l + r) >> 1; if (seg_ids[m] < t0) l = m + 1; else r = m; }
        lo = l;
    }
    {
        int l = lo, r = E;
        while (l < r) { int m = (l + r) >> 1; if (seg_ids[m] < t1) l = m + 1; else r = m; }
        hi = l;
    }
    const int count = hi - lo;

    const int lane = threadIdx.x;   // 0..31, EXEC all-1s (required for WMMA)
    const int half = lane >> 4;     // 0: lanes 0-15, 1: lanes 16-31
    const int col  = lane & 15;     // column within a 16-wide tile

    v2f ones; ones.x = 1.0f; ones.y = 1.0f;   // A-matrix: 16x4 of 1.0
    v8f acc0 = {}, acc1 = {}, acc2 = {}, acc3 = {};

    for (int r = lo; r < hi; r += 4) {
        // VGPR0 of B holds rows {r, r+1} (one per lane half),
        // VGPR1 of B holds rows {r+2, r+3}. Sum over K is invariant to the
        // exact half<->K striping, so this covers all 4 rows exactly once.
        const int  r0   = r + half;
        const int  r1   = r + 2 + half;
        const int  c0   = (r0 < hi) ? r0 : (hi - 1);   // clamp: stay in-bounds
        const int  c1   = (r1 < hi) ? r1 : (hi - 1);   // (loop body => hi > lo)
        const float m0  = (r0 < hi) ? 1.0f : 0.0f;     // branch-free tail mask
        const float m1  = (r1 < hi) ? 1.0f : 0.0f;

        const float* p0 = values + (long)c0 * DCOLS + col;
        const float* p1 = values + (long)c1 * DCOLS + col;

        // prefetch the next chunk of this segment's stream (speculative-safe)
        __builtin_prefetch(p0 + 4 * DCOLS, 0, 1);

        v2f b0, b1, b2, b3;
        b0.x = p0[ 0] * m0;  b0.y = p1[ 0] * m1;   // cols  0..15
        b1.x = p0[16] * m0;  b1.y = p1[16] * m1;   // cols 16..31
        b2.x = p0[32] * m0;  b2.y = p1[32] * m1;   // cols 32..47
        b3.x = p0[48] * m0;  b3.y = p1[48] * m1;   // cols 48..63

        acc0 = __builtin_amdgcn_wmma_f32_16x16x4_f32(false, ones, false, b0,
                                                     (short)0, acc0, false, false);
        acc1 = __builtin_amdgcn_wmma_f32_16x16x4_f32(false, ones, false, b1,
                                                     (short)0, acc1, false, false);
        acc2 = __builtin_amdgcn_wmma_f32_16x16x4_f32(false, ones, false, b2,
                                                     (short)0, acc2, false, false);
        acc3 = __builtin_amdgcn_wmma_f32_16x16x4_f32(false, ones, false, b3,
                                                     (short)0, acc3, false, false);
    }

    // VGPR 0 of C/D (D row 0 in lanes 0-15, row 8 in lanes 16-31) holds the column sums
    // (all D rows are identical because A is all-ones).
    const float inv = 1.0f / (float)((count > 0) ? count : 1);
    float* o = out + (long)seg * DCOLS;
    if (half == 0) {
        o[col +  0] = acc0[0] * inv;
        o[col + 16] = acc1[0] * inv;
    } else {
        o[col + 32] = acc2[0] * inv;
        o[col + 48] = acc3[0] * inv;
    }
}

extern "C" void kernel_launch(void* const* d_in, const int* in_sizes, int n_in,
                              void* d_out, int out_size, void* d_ws, size_t ws_size,
                              hipStream_t stream) {
    const float*     values  = (const float*)d_in[0];
    const long long* seg_ids = (const long long*)d_in[1];
    float*           out     = (float*)d_out;

    const int E = in_sizes[0] / DCOLS;   // 1,250,000
    const int N = out_size   / DCOLS;    // 50,000 segments

    seg_mean_wmma_kernel<<<N, 32, 0, stream>>>(values, seg_ids, out, E, N);
}